// edr_loss_10393820856767
// MI455X (gfx1250) — compile-verified
//
#include <hip/hip_runtime.h>
#include <math.h>

// ---------------------------------------------------------------------------
// EDR loss pipeline for MI455X (gfx1250, wave32, WMMA f32 16x16x4)
// ---------------------------------------------------------------------------

typedef float v2f __attribute__((ext_vector_type(2)));
typedef float v8f __attribute__((ext_vector_type(8)));

#define WMMA4(a, b, c) \
    __builtin_amdgcn_wmma_f32_16x16x4_f32(false, (a), false, (b), (short)0, (c), false, false)

#define V8F_ZERO {0.f,0.f,0.f,0.f,0.f,0.f,0.f,0.f}

// Problem constants
constexpr int BATCH = 128;
constexpr int NFFT  = 96000;       // irfft length
constexpr int N1    = 256;         // Cooley-Tukey factor 1 (DFT-256)
constexpr int N2    = 375;         // Cooley-Tukey factor 2 (DFT-375)
constexpr int N2P   = 384;         // N2 padded to 16-tile multiple
constexpr int KP    = 376;         // N2 padded to multiple of 4 (WMMA K-step)
constexpr int WSZ   = 512;         // STFT window / fft size
constexpr int HOP   = 256;
constexpr int TFR   = 374;         // frames per row: 1 + (96000-512)/256
constexpr int FBIN  = 257;         // onesided bins
constexpr int FBP   = 272;         // bins padded to 16-tile multiple
constexpr int MFR   = BATCH * TFR; // 47872 total frames (= 2992 * 16)

// ------------------------- table-generation kernels -------------------------

__global__ void k_tab_d256(float* __restrict__ dc, float* __restrict__ ds) {
    int i = blockIdx.x * blockDim.x + threadIdx.x;
    if (i >= N1 * N1) return;
    int t1 = i / N1, k1 = i % N1;
    double th = 2.0 * M_PI * (double)(t1 * k1) / (double)N1;  // inverse sign (+i)
    dc[i] = (float)cos(th);
    ds[i] = (float)sin(th);
}

__global__ void k_tab_c375(float* __restrict__ c, float* __restrict__ sn) {
    int i = blockIdx.x * blockDim.x + threadIdx.x;
    if (i >= KP * N2P) return;
    int k2 = i / N2P, t2 = i % N2P;
    float cv = 0.f, sv = 0.f;
    if (k2 < N2 && t2 < N2) {
        double th = 2.0 * M_PI * (double)(k2 * t2) / (double)N2;
        cv = (float)cos(th);
        sv = (float)(-sin(th));   // fold the minus sign of Re{(Ar+iAi)e^{+i..}}
    }
    c[i] = cv; sn[i] = sv;
}

__global__ void k_tab_dft512(float* __restrict__ c, float* __restrict__ s) {
    int i = blockIdx.x * blockDim.x + threadIdx.x;
    if (i >= WSZ * FBP) return;
    int j = i / FBP, f = i % FBP;
    float cv = 0.f, sv = 0.f;
    if (f < FBIN) {
        double th = 2.0 * M_PI * (double)(j * f) / (double)WSZ;
        cv = (float)cos(th);
        sv = (float)sin(th);      // sign irrelevant: squared in power
    }
    c[i] = cv; s[i] = sv;
}

__global__ void k_tab_win(float* __restrict__ w) {
    int j = blockIdx.x * blockDim.x + threadIdx.x;
    if (j >= WSZ) return;
    w[j] = 0.5f * (1.0f - cosf(6.283185307179586f * (float)j / (float)WSZ));
}

__global__ void k_zero_acc(float* __restrict__ acc) {
    acc[0] = 0.f; acc[1] = 0.f;
}

// ------------------- stage 0: Hermitian extend + reorder --------------------
// Xr[b][k1][k2p] = X[b, mirror(k2 + 375*k1)] / 96000 ; zero in pad columns.
__global__ void k_build_xr(const float* __restrict__ X, float* __restrict__ Xr) {
    size_t i = (size_t)blockIdx.x * blockDim.x + threadIdx.x;
    if (i >= (size_t)BATCH * N1 * N2P) return;
    int k2p = (int)(i % N2P);
    size_t r = i / N2P;
    int k1 = (int)(r % N1);
    int b  = (int)(r / N1);
    float v = 0.f;
    if (k2p < N2) {
        int k  = k2p + N2 * k1;                  // 0..95999
        int kk = (k <= NFFT / 2) ? k : NFFT - k; // Hermitian mirror (real spectrum)
        v = X[(size_t)b * NFFT + kk] * (1.0f / (float)NFFT);
    }
    Xr[i] = v;
}

// -------- stage 1: A0 = D256 x Xr (per batch), twiddle in epilogue ----------
// grid (3, 16, 128), block (32, 4): one wave per TWO 16x16 N-tiles (A reused),
// K = 256 in steps of 4.
__global__ void k_stage1(const float* __restrict__ Xr,
                         const float* __restrict__ Dc, const float* __restrict__ Ds,
                         float* __restrict__ Ar, float* __restrict__ Ai) {
    const int b     = blockIdx.z;
    const int tileM = blockIdx.y;
    const int pairN = blockIdx.x * 4 + threadIdx.y;  // 0..11 -> N-tiles 2p, 2p+1
    const int lane  = threadIdx.x;
    const int half  = lane >> 4, l = lane & 15;
    const int m  = tileM * 16 + l;        // t1 row for A frag
    const int n0 = pairN * 32 + l;        // k2 col, tile 0
    const int n1 = n0 + 16;               // k2 col, tile 1
    const float* __restrict__ Xb = Xr + (size_t)b * N1 * N2P;

    v8f c0 = V8F_ZERO, c1 = V8F_ZERO, s0 = V8F_ZERO, s1 = V8F_ZERO;
    for (int k0 = 0; k0 < N1; k0 += 4) {
        const int ka = k0 + 2 * half;
        v2f ac, as, b0, b1;
        ac.x = Dc[m * N1 + ka];  ac.y = Dc[m * N1 + ka + 1];
        as.x = Ds[m * N1 + ka];  as.y = Ds[m * N1 + ka + 1];
        b0.x = Xb[(size_t)ka * N2P + n0];  b0.y = Xb[(size_t)(ka + 1) * N2P + n0];
        b1.x = Xb[(size_t)ka * N2P + n1];  b1.y = Xb[(size_t)(ka + 1) * N2P + n1];
        c0 = WMMA4(ac, b0, c0);
        c1 = WMMA4(ac, b1, c1);
        s0 = WMMA4(as, b0, s0);
        s1 = WMMA4(as, b1, s1);
    }
#pragma unroll
    for (int r = 0; r < 8; ++r) {
        const int row = tileM * 16 + r + 8 * half;   // t1
        const size_t ob = (size_t)b * N1 * N2P + (size_t)row * N2P;
#pragma unroll
        for (int u = 0; u < 2; ++u) {
            const int col = (u == 0) ? n0 : n1;       // k2 (pad cols carry zeros)
            const float a0r = (u == 0) ? c0[r] : c1[r];
            const float a0i = (u == 0) ? s0[r] : s1[r];
            const float ang = 6.283185307179586f * (float)(row * col) / (float)NFFT;
            float sg, cg; sincosf(ang, &sg, &cg);     // e^{+2*pi*i*t1*k2/N}
            Ar[ob + col] = a0r * cg - a0i * sg;
            Ai[ob + col] = a0r * sg + a0i * cg;
        }
    }
}

// -------- stage 2: y[t1 + 256*t2] = Ar x C375 + Ai x (-S375) ----------------
// grid (3, 16, 128), block (32, 4): one wave per TWO N-tiles (Ar/Ai reused);
// K = 376 (padded), N = 384 (padded)
__global__ void k_stage2(const float* __restrict__ Ar, const float* __restrict__ Ai,
                         const float* __restrict__ C375, const float* __restrict__ S375n,
                         float* __restrict__ y) {
    const int b     = blockIdx.z;
    const int tileM = blockIdx.y;
    const int pairN = blockIdx.x * 4 + threadIdx.y;
    const int lane  = threadIdx.x;
    const int half  = lane >> 4, l = lane & 15;
    const int m  = tileM * 16 + l;   // t1
    const int n0 = pairN * 32 + l;   // t2, tile 0
    const int n1 = n0 + 16;          // t2, tile 1
    const float* __restrict__ Arb = Ar + (size_t)b * N1 * N2P;
    const float* __restrict__ Aib = Ai + (size_t)b * N1 * N2P;

    v8f a0 = V8F_ZERO, a1 = V8F_ZERO;
    for (int k0 = 0; k0 < KP; k0 += 4) {
        const int ka = k0 + 2 * half;
        v2f ar, ai, bc0, bs0, bc1, bs1;
        ar.x = Arb[(size_t)m * N2P + ka];  ar.y = Arb[(size_t)m * N2P + ka + 1];
        ai.x = Aib[(size_t)m * N2P + ka];  ai.y = Aib[(size_t)m * N2P + ka + 1];
        bc0.x = C375[(size_t)ka * N2P + n0];  bc0.y = C375[(size_t)(ka + 1) * N2P + n0];
        bs0.x = S375n[(size_t)ka * N2P + n0]; bs0.y = S375n[(size_t)(ka + 1) * N2P + n0];
        bc1.x = C375[(size_t)ka * N2P + n1];  bc1.y = C375[(size_t)(ka + 1) * N2P + n1];
        bs1.x = S375n[(size_t)ka * N2P + n1]; bs1.y = S375n[(size_t)(ka + 1) * N2P + n1];
        a0 = WMMA4(ar, bc0, a0);
        a0 = WMMA4(ai, bs0, a0);
        a1 = WMMA4(ar, bc1, a1);
        a1 = WMMA4(ai, bs1, a1);
    }
#pragma unroll
    for (int r = 0; r < 8; ++r) {
        const int row = tileM * 16 + r + 8 * half;   // t1
        if (n0 < N2) y[(size_t)b * NFFT + row + N1 * n0] = a0[r];
        if (n1 < N2) y[(size_t)b * NFFT + row + N1 * n1] = a1[r];
    }
}

// -------- stage 3: STFT power via GEMM against DFT-512 basis ----------------
// grid (17, 374), block (32, 4): one wave per TWO M-tiles (B frags reused);
// M = 47872 frames, N = 272 (padded), K = 512
__global__ void k_stft(const float* __restrict__ y,
                       const float* __restrict__ dftc, const float* __restrict__ dfts,
                       const float* __restrict__ win, float* __restrict__ pw) {
    const int tileN = blockIdx.x;
    const int pairM = blockIdx.y * 4 + threadIdx.y;  // 0..1495 -> M-tiles 2q, 2q+1
    const int lane  = threadIdx.x;
    const int half  = lane >> 4, l = lane & 15;
    const int m0 = pairM * 32 + l;        // frame id, tile 0
    const int m1 = m0 + 16;               // frame id, tile 1
    const float* __restrict__ yr0 = y + (size_t)(m0 / TFR) * NFFT + (size_t)(m0 % TFR) * HOP;
    const float* __restrict__ yr1 = y + (size_t)(m1 / TFR) * NFFT + (size_t)(m1 % TFR) * HOP;
    const int n = tileN * 16 + l;

    v8f c0 = V8F_ZERO, c1 = V8F_ZERO, s0 = V8F_ZERO, s1 = V8F_ZERO;
    for (int k0 = 0; k0 < WSZ; k0 += 4) {
        const int ka = k0 + 2 * half;
        const float w0 = win[ka], w1 = win[ka + 1];
        v2f a0, a1, bc, bs;
        a0.x = yr0[ka] * w0;  a0.y = yr0[ka + 1] * w1;
        a1.x = yr1[ka] * w0;  a1.y = yr1[ka + 1] * w1;
        bc.x = dftc[(size_t)ka * FBP + n];  bc.y = dftc[(size_t)(ka + 1) * FBP + n];
        bs.x = dfts[(size_t)ka * FBP + n];  bs.y = dfts[(size_t)(ka + 1) * FBP + n];
        c0 = WMMA4(a0, bc, c0);
        s0 = WMMA4(a0, bs, s0);
        c1 = WMMA4(a1, bc, c1);
        s1 = WMMA4(a1, bs, s1);
    }
    if (n < FBIN) {
#pragma unroll
        for (int r = 0; r < 8; ++r) {
            const int rr = r + 8 * half;
            const int row0 = pairM * 32 + rr;        // frame id = b*374 + t
            const int row1 = row0 + 16;
            pw[(size_t)row0 * FBIN + n] = c0[r] * c0[r] + s0[r] * s0[r];
            pw[(size_t)row1 * FBIN + n] = c1[r] * c1[r] + s1[r] * s1[r];
        }
    }
}

// -------- stage 4: suffix-sum EDR in dB + |diff| reduction ------------------
// One workgroup per batch row; threads span frequency bins so every time-step
// load is contiguous across lanes (coalesced; ~100 MB total instead of ~3 GB).
__global__ void k_edr_reduce(const float* __restrict__ pt, const float* __restrict__ pa,
                             float* __restrict__ acc) {
    const int b = blockIdx.x;
    const int f = threadIdx.x;              // blockDim.x == 512, active f < 257
    float num = 0.f, den = 0.f;
    if (f < FBIN) {
        const float* baseT = pt + (size_t)b * TFR * FBIN + f;
        const float* baseA = pa + (size_t)b * TFR * FBIN + f;
        float st = 0.f, sa = 0.f;
        for (int t = TFR - 1; t >= 0; --t) {
            st += baseT[(size_t)t * FBIN];
            sa += baseA[(size_t)t * FBIN];
            const float et = 10.f * log10f(st);
            const float ea = 10.f * log10f(sa);
            num += fabsf(et - ea);
            den += fabsf(et);
        }
    }
    __shared__ float sN[512], sD[512];
    const int tid = threadIdx.x;
    sN[tid] = num; sD[tid] = den;
    __syncthreads();
    for (int s = 256; s > 0; s >>= 1) {
        if (tid < s) { sN[tid] += sN[tid + s]; sD[tid] += sD[tid + s]; }
        __syncthreads();
    }
    if (tid == 0) {
        atomicAdd(&acc[0], sN[0]);
        atomicAdd(&acc[1], sD[0]);
    }
}

__global__ void k_finalize(const float* __restrict__ acc, float* __restrict__ out) {
    out[0] = acc[0] / acc[1];
}

// ------------------------------ host launcher -------------------------------

extern "C" void kernel_launch(void* const* d_in, const int* in_sizes, int n_in,
                              void* d_out, int out_size, void* d_ws, size_t ws_size,
                              hipStream_t stream) {
    // workspace carving (float offsets)
    constexpr size_t SZ_D256 = (size_t)N1 * N1;          // 65536
    constexpr size_t SZ_C375 = (size_t)KP * N2P;         // 144384
    constexpr size_t SZ_DFT  = (size_t)WSZ * FBP;        // 139264
    constexpr size_t SZ_BIG  = (size_t)BATCH * N1 * N2P; // 12582912
    constexpr size_t SZ_POW  = (size_t)BATCH * TFR * FBIN;

    size_t o = 0;
    const size_t O_D256C = o; o += SZ_D256;
    const size_t O_D256S = o; o += SZ_D256;
    const size_t O_C375  = o; o += SZ_C375;
    const size_t O_S375  = o; o += SZ_C375;
    const size_t O_DFTC  = o; o += SZ_DFT;
    const size_t O_DFTS  = o; o += SZ_DFT;
    const size_t O_WIN   = o; o += WSZ;
    const size_t O_ACC   = o; o += 16;
    const size_t O_XR    = o; o += SZ_BIG;   // reused as y after stage 1 consumes it
    const size_t O_AR    = o; o += SZ_BIG;
    const size_t O_AI    = o; o += SZ_BIG;
    const size_t O_PT    = o; o += SZ_POW;
    const size_t O_PA    = o; o += SZ_POW;
    if (ws_size < o * sizeof(float)) return;  // scratch too small: nothing safe to do

    float* w = (float*)d_ws;
    float* d256c = w + O_D256C; float* d256s = w + O_D256S;
    float* c375  = w + O_C375;  float* s375n = w + O_S375;
    float* dftc  = w + O_DFTC;  float* dfts  = w + O_DFTS;
    float* win   = w + O_WIN;   float* acc   = w + O_ACC;
    float* xr    = w + O_XR;    float* yy    = w + O_XR;   // alias (sequenced)
    float* Ar    = w + O_AR;    float* Ai    = w + O_AI;
    float* powbuf[2] = { w + O_PT, w + O_PA };

    // tables + accumulator init (cheap, deterministic, graph-safe)
    k_tab_d256  <<<(SZ_D256 + 255) / 256, 256, 0, stream>>>(d256c, d256s);
    k_tab_c375  <<<(SZ_C375 + 255) / 256, 256, 0, stream>>>(c375, s375n);
    k_tab_dft512<<<(SZ_DFT  + 255) / 256, 256, 0, stream>>>(dftc, dfts);
    k_tab_win   <<<2, 256, 0, stream>>>(win);
    k_zero_acc  <<<1, 1, 0, stream>>>(acc);

    const dim3 blkW(32, 4);
    const dim3 grdS12(3, 16, BATCH);               // 12 N-tile pairs, 16 M-tiles
    const dim3 grdSTFT(FBP / 16, (MFR / 32) / 4);  // (17, 374): 1496 M-tile pairs

    for (int sig = 0; sig < 2; ++sig) {
        const float* X = (const float*)d_in[sig];
        k_build_xr<<<(SZ_BIG + 255) / 256, 256, 0, stream>>>(X, xr);
        k_stage1  <<<grdS12, blkW, 0, stream>>>(xr, d256c, d256s, Ar, Ai);
        k_stage2  <<<grdS12, blkW, 0, stream>>>(Ar, Ai, c375, s375n, yy);
        k_stft    <<<grdSTFT, blkW, 0, stream>>>(yy, dftc, dfts, win, powbuf[sig]);
    }

    k_edr_reduce<<<BATCH, 512, 0, stream>>>(powbuf[0], powbuf[1], acc);
    k_finalize  <<<1, 1, 0, stream>>>(acc, (float*)d_out);
}